// ReceiverGAT_20693152432917
// MI455X (gfx1250) — compile-verified
//
#include <hip/hip_runtime.h>
#include <math.h>

typedef __attribute__((ext_vector_type(2))) float v2f;
typedef __attribute__((ext_vector_type(8))) float v8f;

#define HEADS 4
#define EMB 16
#define OUTD 64      // HEADS*EMB
#define INFEAT 4
#define HIDDEN 256
#define NEG_SLOPE 0.2f

// ---- monotone float<->uint key (for atomic segment-max; key(x)>0 for all x>= -inf, so 0-init works) ----
__device__ __forceinline__ unsigned fkey(float f) {
    unsigned u = __float_as_uint(f);
    return (u & 0x80000000u) ? ~u : (u | 0x80000000u);
}
__device__ __forceinline__ float funkey(unsigned k) {
    unsigned u = (k & 0x80000000u) ? (k ^ 0x80000000u) : ~k;
    return __uint_as_float(u);
}

// ---- zero fill ----
__global__ void zero_kernel(unsigned* __restrict__ p, long long n) {
    long long i = (long long)blockIdx.x * blockDim.x + threadIdx.x;
    if (i < n) p[i] = 0u;
}

// ---- h = x @ gat_w ; a_s = <h,att_src> ; a_d = <h,att_dst> per head ----
__global__ void node_kernel(const float* __restrict__ x, const float* __restrict__ gat_w,
                            const float* __restrict__ att_src, const float* __restrict__ att_dst,
                            float* __restrict__ h, float* __restrict__ a_s, float* __restrict__ a_d,
                            int N) {
    __shared__ float w[INFEAT * OUTD];
    __shared__ float asv[OUTD];
    __shared__ float adv[OUTD];
    for (int i = threadIdx.x; i < INFEAT * OUTD; i += blockDim.x) w[i] = gat_w[i];
    for (int i = threadIdx.x; i < OUTD; i += blockDim.x) { asv[i] = att_src[i]; adv[i] = att_dst[i]; }
    __syncthreads();
    int n = blockIdx.x * blockDim.x + threadIdx.x;
    if (n >= N) return;
    float x0 = x[n * 4 + 0], x1 = x[n * 4 + 1], x2 = x[n * 4 + 2], x3 = x[n * 4 + 3];
    float sacc[HEADS] = {0.f, 0.f, 0.f, 0.f};
    float dacc[HEADS] = {0.f, 0.f, 0.f, 0.f};
    #pragma unroll
    for (int j = 0; j < OUTD; ++j) {
        float hj = x0 * w[j] + x1 * w[OUTD + j] + x2 * w[2 * OUTD + j] + x3 * w[3 * OUTD + j];
        h[(size_t)n * OUTD + j] = hj;
        int hd = j >> 4;
        sacc[hd] += hj * asv[j];
        dacc[hd] += hj * adv[j];
    }
    #pragma unroll
    for (int hd = 0; hd < HEADS; ++hd) {
        a_s[(size_t)n * HEADS + hd] = sacc[hd];
        a_d[(size_t)n * HEADS + hd] = dacc[hd];
    }
}

// ---- pass 1: e = leaky_relu(a_s[src]+a_d[dst]); segment max via atomic ordered-key max ----
__global__ void edge_score_kernel(const int* __restrict__ src, const int* __restrict__ dst,
                                  const float* __restrict__ a_s, const float* __restrict__ a_d,
                                  float* __restrict__ ebuf, unsigned* __restrict__ mkey, int E) {
    int e = blockIdx.x * blockDim.x + threadIdx.x;
    if (e >= E) return;
    int s = src[e], d = dst[e];
    float4 as4 = *(const float4*)(a_s + (size_t)s * HEADS);
    float4 ad4 = *(const float4*)(a_d + (size_t)d * HEADS);
    float v[HEADS] = {as4.x + ad4.x, as4.y + ad4.y, as4.z + ad4.z, as4.w + ad4.w};
    float4 out;
    float* op = &out.x;
    #pragma unroll
    for (int k = 0; k < HEADS; ++k) {
        float t = v[k];
        t = t > 0.f ? t : NEG_SLOPE * t;
        op[k] = t;
        atomicMax(&mkey[(size_t)d * HEADS + k], fkey(t));
    }
    *(float4*)(ebuf + (size_t)e * HEADS) = out;
}

// ---- pass 2: ex = exp(e - m[dst]); denom = segment_sum(ex) ----
__global__ void edge_exp_kernel(const int* __restrict__ dst, float* __restrict__ ebuf,
                                const unsigned* __restrict__ mkey, float* __restrict__ denom, int E) {
    int e = blockIdx.x * blockDim.x + threadIdx.x;
    if (e >= E) return;
    int d = dst[e];
    float4 ev = *(float4*)(ebuf + (size_t)e * HEADS);
    float* ep = &ev.x;
    #pragma unroll
    for (int k = 0; k < HEADS; ++k) {
        float m = funkey(mkey[(size_t)d * HEADS + k]);
        float ex = expf(ep[k] - m);
        ep[k] = ex;
        atomicAdd(&denom[(size_t)d * HEADS + k], ex);
    }
    *(float4*)(ebuf + (size_t)e * HEADS) = ev;
}

// ---- pass 3: agg[dst] += alpha * h[src] ; one thread per (edge, head) ----
__global__ void edge_agg_kernel(const int* __restrict__ src, const int* __restrict__ dst,
                                const float* __restrict__ ebuf, const float* __restrict__ denom,
                                const float* __restrict__ h, float* __restrict__ agg, int E) {
    long long t = (long long)blockIdx.x * blockDim.x + threadIdx.x;
    if (t >= (long long)E * HEADS) return;
    int e = (int)(t >> 2), k = (int)(t & 3);
    int s = src[e], d = dst[e];
    const float* hs = h + (size_t)s * OUTD + k * EMB;
    __builtin_prefetch(hs, 0, 3);                 // global_prefetch_b8: warm the gather row
    float alpha = ebuf[(size_t)e * HEADS + k] / (denom[(size_t)d * HEADS + k] + 1e-16f);
    float* ag = agg + (size_t)d * OUTD + k * EMB;
    #pragma unroll
    for (int f = 0; f < EMB; ++f) atomicAdd(&ag[f], alpha * hs[f]);
}

// ---- h_out = agg + gat_b (in place) ----
__global__ void bias_kernel(float* __restrict__ agg, const float* __restrict__ gat_b, long long n) {
    long long i = (long long)blockIdx.x * blockDim.x + threadIdx.x;
    if (i < n) agg[i] += gat_b[i & (OUTD - 1)];
}

// ---- msg_emb = message @ fc_w + fc_b  via V_WMMA_F32_16X16X4_F32 (one wave per 16x16 tile) ----
__global__ void fc_wmma_kernel(const float* __restrict__ message, const float* __restrict__ fc_w,
                               const float* __restrict__ fc_b, float* __restrict__ msg_emb, int B) {
    int wave = (blockIdx.x * blockDim.x + threadIdx.x) >> 5;
    int lane = threadIdx.x & 31;
    const int ntn = OUTD / 16;               // 4 tiles in N
    int ntiles = (B / 16) * ntn;
    if (wave >= ntiles) return;              // wave-uniform: EXEC stays all-ones for WMMA
    int b0 = (wave / ntn) * 16;
    int n0 = (wave % ntn) * 16;
    int r = lane & 15, hi = lane >> 4;
    const float* arow = message + (size_t)(b0 + r) * HIDDEN;
    v8f acc = {};
    for (int i = 0; i < HIDDEN / 4; ++i) {
        int k = 4 * i + 2 * hi;
        v2f a = *(const v2f*)(arow + k);                       // A[r][k], A[r][k+1]
        v2f b;
        b.x = fc_w[(size_t)k * OUTD + n0 + r];                 // B[k][n]
        b.y = fc_w[(size_t)(k + 1) * OUTD + n0 + r];
        acc = __builtin_amdgcn_wmma_f32_16x16x4_f32(false, a, false, b, (short)0, acc, false, false);
    }
    float bias = fc_b[n0 + r];
    #pragma unroll
    for (int j = 0; j < 8; ++j)
        msg_emb[(size_t)(b0 + j + 8 * hi) * OUTD + n0 + r] = acc[j] + bias;
}

// ---- logits[b,n] = sum_k h_out[n,k]*msg_emb[b,k] ----
// 2x2 register-blocked: each wave owns two 16-row b-tiles and two 16-col n-tiles;
// every A/B fragment feeds two WMMAs -> 8 FLOP/byte of operand traffic, 64 wmma/wave.
__global__ void logits_wmma_kernel(const float* __restrict__ h_out, const float* __restrict__ msg_emb,
                                   float* __restrict__ logits, int N, int B) {
    int wave = (blockIdx.x * blockDim.x + threadIdx.x) >> 5;
    int lane = threadIdx.x & 31;
    int ntn = (N + 15) / 16;                  // 16-col n-tiles
    int nnp = (ntn + 1) >> 1;                 // n-tile pairs
    int nbp = B >> 5;                         // b-tile pairs (B multiple of 32)
    long long npairs = (long long)nbp * nnp;
    if (wave >= npairs) return;               // wave-uniform guard: EXEC all-ones for WMMA
    int bp = wave / nnp, np = wave % nnp;
    int b0 = bp * 32;
    int n0 = np * 32;
    int r = lane & 15, hi = lane >> 4;
    int nr0 = n0 + r;      if (nr0 > N - 1) nr0 = N - 1;   // clamp loads (keep all lanes active)
    int nr1 = n0 + 16 + r; if (nr1 > N - 1) nr1 = N - 1;
    const float* arow0 = msg_emb + (size_t)(b0 + r) * OUTD;
    const float* arow1 = msg_emb + (size_t)(b0 + 16 + r) * OUTD;
    const float* brow0 = h_out + (size_t)nr0 * OUTD;
    const float* brow1 = h_out + (size_t)nr1 * OUTD;
    v8f acc00 = {}, acc01 = {}, acc10 = {}, acc11 = {};
    #pragma unroll
    for (int i = 0; i < OUTD / 4; ++i) {
        int k = 4 * i + 2 * hi;
        v2f a0 = *(const v2f*)(arow0 + k);    // A row fragments (b-tiles)
        v2f a1 = *(const v2f*)(arow1 + k);
        v2f bb0 = *(const v2f*)(brow0 + k);   // B = h_out^T fragments (n-tiles), contiguous 8B
        v2f bb1 = *(const v2f*)(brow1 + k);
        acc00 = __builtin_amdgcn_wmma_f32_16x16x4_f32(false, a0, false, bb0, (short)0, acc00, false, false);
        acc01 = __builtin_amdgcn_wmma_f32_16x16x4_f32(false, a0, false, bb1, (short)0, acc01, false, false);
        acc10 = __builtin_amdgcn_wmma_f32_16x16x4_f32(false, a1, false, bb0, (short)0, acc10, false, false);
        acc11 = __builtin_amdgcn_wmma_f32_16x16x4_f32(false, a1, false, bb1, (short)0, acc11, false, false);
    }
    bool ok0 = (n0 + r) < N, ok1 = (n0 + 16 + r) < N;
    if (ok0) {
        #pragma unroll
        for (int j = 0; j < 8; ++j) {
            logits[(size_t)(b0 + j + 8 * hi) * N + n0 + r]      = acc00[j];
            logits[(size_t)(b0 + 16 + j + 8 * hi) * N + n0 + r] = acc10[j];
        }
    }
    if (ok1) {
        #pragma unroll
        for (int j = 0; j < 8; ++j) {
            logits[(size_t)(b0 + j + 8 * hi) * N + n0 + 16 + r]      = acc01[j];
            logits[(size_t)(b0 + 16 + j + 8 * hi) * N + n0 + 16 + r] = acc11[j];
        }
    }
}

// ---- row-wise log_softmax over N, in place; one block per batch row ----
__global__ void logsoftmax_kernel(float* __restrict__ logits, int N) {
    __shared__ float red[256];
    float* row = logits + (size_t)blockIdx.x * N;
    float mx = -INFINITY;
    for (int i = threadIdx.x; i < N; i += blockDim.x) mx = fmaxf(mx, row[i]);
    red[threadIdx.x] = mx;
    __syncthreads();
    for (int s = 128; s > 0; s >>= 1) {
        if ((int)threadIdx.x < s) red[threadIdx.x] = fmaxf(red[threadIdx.x], red[threadIdx.x + s]);
        __syncthreads();
    }
    mx = red[0];
    __syncthreads();
    float sum = 0.f;
    for (int i = threadIdx.x; i < N; i += blockDim.x) sum += expf(row[i] - mx);
    red[threadIdx.x] = sum;
    __syncthreads();
    for (int s = 128; s > 0; s >>= 1) {
        if ((int)threadIdx.x < s) red[threadIdx.x] += red[threadIdx.x + s];
        __syncthreads();
    }
    float lse = mx + logf(red[0]);
    __syncthreads();
    for (int i = threadIdx.x; i < N; i += blockDim.x) row[i] -= lse;
}

extern "C" void kernel_launch(void* const* d_in, const int* in_sizes, int n_in,
                              void* d_out, int out_size, void* d_ws, size_t ws_size,
                              hipStream_t stream) {
    const float* message = (const float*)d_in[0];
    const float* x       = (const float*)d_in[1];
    const int*   eidx    = (const int*)d_in[2];
    const float* gat_w   = (const float*)d_in[3];
    const float* att_src = (const float*)d_in[4];
    const float* att_dst = (const float*)d_in[5];
    const float* gat_b   = (const float*)d_in[6];
    const float* fc_w    = (const float*)d_in[7];
    const float* fc_b    = (const float*)d_in[8];

    int N = in_sizes[1] / INFEAT;       // 50000
    int E = in_sizes[2] / 2;            // 1600000
    int B = in_sizes[0] / HIDDEN;       // 1024
    const int* src = eidx;
    const int* dst = eidx + E;

    // workspace layout (floats)
    float* ws = (float*)d_ws;
    size_t o = 0;
    float*    h       = ws + o;               o += (size_t)N * OUTD;
    float*    a_s     = ws + o;               o += (size_t)N * HEADS;
    float*    a_d     = ws + o;               o += (size_t)N * HEADS;
    unsigned* mkey    = (unsigned*)(ws + o);  o += (size_t)N * HEADS;
    float*    denom   = ws + o;               o += (size_t)N * HEADS;
    float*    ebuf    = ws + o;               o += (size_t)E * HEADS;
    float*    agg     = ws + o;               o += (size_t)N * OUTD;
    float*    msg_emb = ws + o;               o += (size_t)B * OUTD;

    float* logits = (float*)d_out;
    const int T = 256;

    // zero mkey + denom (contiguous) and agg
    {
        long long n1 = (long long)N * HEADS * 2;
        zero_kernel<<<(unsigned)((n1 + T - 1) / T), T, 0, stream>>>(mkey, n1);
        long long n2 = (long long)N * OUTD;
        zero_kernel<<<(unsigned)((n2 + T - 1) / T), T, 0, stream>>>((unsigned*)agg, n2);
    }

    node_kernel<<<(N + T - 1) / T, T, 0, stream>>>(x, gat_w, att_src, att_dst, h, a_s, a_d, N);
    edge_score_kernel<<<(E + T - 1) / T, T, 0, stream>>>(src, dst, a_s, a_d, ebuf, mkey, E);
    edge_exp_kernel<<<(E + T - 1) / T, T, 0, stream>>>(dst, ebuf, mkey, denom, E);
    {
        long long n = (long long)E * HEADS;
        edge_agg_kernel<<<(unsigned)((n + T - 1) / T), T, 0, stream>>>(src, dst, ebuf, denom, h, agg, E);
    }
    {
        long long n = (long long)N * OUTD;
        bias_kernel<<<(unsigned)((n + T - 1) / T), T, 0, stream>>>(agg, gat_b, n);
    }
    {
        int tiles = (B / 16) * (OUTD / 16);
        long long thr = (long long)tiles * 32;
        fc_wmma_kernel<<<(unsigned)((thr + T - 1) / T), T, 0, stream>>>(message, fc_w, fc_b, msg_emb, B);
    }
    {
        int ntn = (N + 15) / 16;
        long long pairs = (long long)(B >> 5) * ((ntn + 1) >> 1);
        long long thr = pairs * 32;
        logits_wmma_kernel<<<(unsigned)((thr + T - 1) / T), T, 0, stream>>>(agg, msg_emb, logits, N, B);
    }
    logsoftmax_kernel<<<B, T, 0, stream>>>(logits, N);
}